// Diff_Label01_Loss_6854767804505
// MI455X (gfx1250) — compile-verified
//
#include <hip/hip_runtime.h>

#define NROWS 8192
#define NCOLS 4096
#define EPSV  1e-8f

typedef float v2f __attribute__((ext_vector_type(2)));
typedef float v8f __attribute__((ext_vector_type(8)));

// ws layout (floats): [0,4096) colsum | [4096,8192) mean | [8192,8200) scalars
// scalars: [0]=n0 [1]=n1 [2]=||mean||^2 [3]=sim_sum [4]=differ_sum

__global__ void k_zero(float* ws) {
    int i = blockIdx.x * blockDim.x + threadIdx.x;
    if (i < NCOLS) ws[i] = 0.0f;                  // colsum
    if (i < 8)     ws[2 * NCOLS + i] = 0.0f;      // scalars
}

__global__ void k_count(const float* __restrict__ labels, float* __restrict__ scal) {
    __shared__ float s0[256];
    __shared__ float s1[256];
    int i  = blockIdx.x * 256 + threadIdx.x;
    float l0 = labels[2 * i];
    float l1 = labels[2 * i + 1];
    float m0 = (l0 >= l1) ? 1.0f : 0.0f;          // argmax -> first max wins
    s0[threadIdx.x] = m0;
    s1[threadIdx.x] = 1.0f - m0;
    __syncthreads();
    for (int off = 128; off > 0; off >>= 1) {
        if (threadIdx.x < off) {
            s0[threadIdx.x] += s0[threadIdx.x + off];
            s1[threadIdx.x] += s1[threadIdx.x + off];
        }
        __syncthreads();
    }
    if (threadIdx.x == 0) {
        atomicAdd(&scal[0], s0[0]);
        atomicAdd(&scal[1], s1[0]);
    }
}

// Masked column sum. Block = 256 threads x float4 -> 1024 contiguous columns.
// Grid: (NCOLS/1024, NROWS/128). Row-chunk masks staged in LDS once.
#define CS_ROWS 128
__global__ void k_colsum(const float* __restrict__ labels,
                         const float* __restrict__ datas,
                         float* __restrict__ colsum) {
    __shared__ float smask[CS_ROWS];
    int col4 = blockIdx.x * 256 + threadIdx.x;    // float4 index
    int r0   = blockIdx.y * CS_ROWS;

    if (threadIdx.x < CS_ROWS) {
        int r = r0 + threadIdx.x;
        smask[threadIdx.x] = (labels[2 * r] >= labels[2 * r + 1]) ? 1.0f : 0.0f;
    }
    __syncthreads();

    float4 acc = {0.0f, 0.0f, 0.0f, 0.0f};
    const float4* dp = (const float4*)datas;
#pragma unroll 4
    for (int i = 0; i < CS_ROWS; ++i) {
        if (smask[i] > 0.0f) {                    // uniform across block
            float4 v = dp[(size_t)(r0 + i) * (NCOLS / 4) + col4];
            acc.x += v.x; acc.y += v.y; acc.z += v.z; acc.w += v.w;
        }
    }
    float* cs = colsum + 4 * col4;
    atomicAdd(&cs[0], acc.x);
    atomicAdd(&cs[1], acc.y);
    atomicAdd(&cs[2], acc.z);
    atomicAdd(&cs[3], acc.w);
}

__global__ void k_mean(const float* __restrict__ colsum,
                       float* __restrict__ mean,
                       float* __restrict__ scal) {
    __shared__ float red[256];
    int d = blockIdx.x * 256 + threadIdx.x;
    float inv = 1.0f / fmaxf(scal[0], 1.0f);
    float mv  = colsum[d] * inv;
    mean[d] = mv;
    red[threadIdx.x] = mv * mv;
    __syncthreads();
    for (int off = 128; off > 0; off >>= 1) {
        if (threadIdx.x < off) red[threadIdx.x] += red[threadIdx.x + off];
        __syncthreads();
    }
    if (threadIdx.x == 0) atomicAdd(&scal[2], red[0]);
}

// One wave (32 lanes) per 16 rows. WMMA f32 16x16x4:
//   c_dot  accumulates D[m][n] = dot(row_m, mean)  (all n equal, B replicated)
//   c_gram accumulates Gram;   diag = ||row_m||^2  (B frag == A frag registers)
__global__ void __launch_bounds__(32) k_wmma(const float* __restrict__ labels,
                                             const float* __restrict__ datas,
                                             const float* __restrict__ mean,
                                             float* __restrict__ scal) {
    __shared__ float lds_dot[256];
    __shared__ float lds_gram[256];
    __shared__ float red_sim[32];
    __shared__ float red_dif[32];

    const int lane = threadIdx.x;
    const int m = lane & 15;
    const int e = lane >> 4;
    const int rowbase = blockIdx.x * 16;

    // slot(j,e) <-> column 8*it + 4e + j: fold lane-group e into the base
    // pointer so the loop trip count is uniform (scalar branch, real unroll).
    const float4* dptr = (const float4*)(datas + (size_t)(rowbase + m) * NCOLS) + e;
    const float4* mptr = (const float4*)mean + e;

    v8f c_dot = {};
    v8f c_gram = {};

#pragma unroll 4
    for (int it = 0; it < NCOLS / 8; ++it) {
        float4 d4 = dptr[2 * it];
        float4 m4 = mptr[2 * it];
        v2f a0 = {d4.x, d4.y};
        v2f a1 = {d4.z, d4.w};
        v2f b0 = {m4.x, m4.y};
        v2f b1 = {m4.z, m4.w};
        c_dot  = __builtin_amdgcn_wmma_f32_16x16x4_f32(false, a0, false, b0, (short)0, c_dot,  false, false);
        c_gram = __builtin_amdgcn_wmma_f32_16x16x4_f32(false, a0, false, a0, (short)0, c_gram, false, false);
        c_dot  = __builtin_amdgcn_wmma_f32_16x16x4_f32(false, a1, false, b1, (short)0, c_dot,  false, false);
        c_gram = __builtin_amdgcn_wmma_f32_16x16x4_f32(false, a1, false, a1, (short)0, c_gram, false, false);
    }

    // C/D layout: vgpr v, lanes 0-15 -> (M=v, N=lane); lanes 16-31 -> (M=v+8, N=lane-16)
#pragma unroll
    for (int v = 0; v < 8; ++v) {
        int M = v + 8 * e;
        lds_dot[M * 16 + m]  = c_dot[v];
        lds_gram[M * 16 + m] = c_gram[v];
    }
    __syncthreads();

    float simc = 0.0f, difc = 0.0f;
    if (lane < 16) {
        int r = rowbase + lane;
        float dot   = lds_dot[lane * 16 + 0];       // any column; take N=0
        float nrm2  = lds_gram[lane * 16 + lane];   // Gram diagonal
        float xnorm = fmaxf(sqrtf(nrm2), EPSV);
        float mnorm = fmaxf(sqrtf(scal[2]), EPSV);
        float cosv  = dot / (xnorm * mnorm);
        float ac    = fabsf(cosv);
        float l0 = labels[2 * r];
        float l1 = labels[2 * r + 1];
        float m0 = (l0 >= l1) ? 1.0f : 0.0f;
        simc = m0 * (1.0f - ac);
        difc = (1.0f - m0) * ac;
    }
    red_sim[lane] = simc;
    red_dif[lane] = difc;
    __syncthreads();
    if (lane == 0) {
        float s = 0.0f, d = 0.0f;
        for (int i = 0; i < 16; ++i) { s += red_sim[i]; d += red_dif[i]; }
        atomicAdd(&scal[3], s);
        atomicAdd(&scal[4], d);
    }
}

__global__ void k_final(const float* __restrict__ scal, float* __restrict__ out) {
    float n0 = scal[0];
    float n1 = scal[1];
    float sim = scal[3] / fmaxf(n0, 1.0f);
    float dif = scal[4] / fmaxf(n1, 1.0f);
    sim = (n0 > 0.0f) ? sim : 0.0f;
    dif = ((n0 > 0.0f) && (n1 > 0.0f)) ? dif : 0.0f;
    out[0] = sim + dif;
    out[1] = sim;
    out[2] = dif;
}

extern "C" void kernel_launch(void* const* d_in, const int* in_sizes, int n_in,
                              void* d_out, int out_size, void* d_ws, size_t ws_size,
                              hipStream_t stream) {
    const float* labels = (const float*)d_in[0];   // [8192, 2]
    const float* datas  = (const float*)d_in[1];   // [8192, 4096]
    float* out = (float*)d_out;                    // [3]
    float* ws  = (float*)d_ws;

    float* colsum = ws;
    float* mean   = ws + NCOLS;
    float* scal   = ws + 2 * NCOLS;

    k_zero<<<16, 256, 0, stream>>>(ws);
    k_count<<<NROWS / 256, 256, 0, stream>>>(labels, scal);
    dim3 g2(NCOLS / 1024, NROWS / CS_ROWS);
    k_colsum<<<g2, 256, 0, stream>>>(labels, datas, colsum);
    k_mean<<<NCOLS / 256, 256, 0, stream>>>(colsum, mean, scal);
    k_wmma<<<NROWS / 16, 32, 0, stream>>>(labels, datas, mean, scal);
    k_final<<<1, 1, 0, stream>>>(scal, out);
}